// DynamicSpectralTilt_IIR_50672024158735
// MI455X (gfx1250) — compile-verified
//
#include <hip/hip_runtime.h>
#include <math.h>

// ---------------- problem constants ----------------
constexpr int BATCH   = 16;
constexpr int T       = 524288;
constexpr int NCH     = 64;                 // hidden channels
constexpr int CHUNK   = 4096;               // samples per block
constexpr int THREADS = 256;                // 8 waves (wave32)
constexpr int NWAVES  = THREADS / 32;
constexpr int SPT     = CHUNK / THREADS;    // 16 samples per thread (serial segment)
constexpr int CPR     = T / CHUNK;          // 128 chunks per row
constexpr int NCHUNK  = BATCH * CPR;        // 2048 blocks
constexpr int F4_PER_BUF    = CHUNK / 4;    // 1024 float4 slots per staged array
constexpr int F4_PER_THREAD = F4_PER_BUF / THREADS; // 4

// workspace layout (floats); total < 32 KB
constexpr int WS_R     = 0;                  // 64 sorted breakpoints
constexpr int WS_MQ    = 64;                 // 65 * (M,Q) pairs = 130 floats
constexpr int WS_AGG   = 256;                // 2*NCHUNK chunk aggregates (A,U)
constexpr int WS_CARRY = WS_AGG + 2*NCHUNK;  // NCHUNK chunk carry-in values

constexpr float KSCALE = 8.3333333333333339e-4f; // 10 / nyquist = 10/12000
constexpr float LOG2E  = 1.4426950408889634f;

// ---------------- CDNA5 async LDS data movement (per-pass guarded) ----------------
// Device-pass existence of these builtins was proven by the round-1 gfx1250
// diagnostic (clang type-checked the (v4i AS1*, v4i AS3*, Ii, Ii) signature).
// Guards must be evaluated per compilation pass: the host pass lacks them.
#if defined(__has_builtin)
#  if __has_builtin(__builtin_amdgcn_global_load_async_to_lds_b128)
#    define HAVE_ASYNC_LOAD 1
#  endif
#  if __has_builtin(__builtin_amdgcn_global_store_async_from_lds_b128)
#    define HAVE_ASYNC_STORE 1
#  endif
#  if __has_builtin(__builtin_amdgcn_s_wait_asynccnt)
#    define HAVE_WAIT_ASYNC 1
#  endif
#endif
#ifndef HAVE_ASYNC_LOAD
#define HAVE_ASYNC_LOAD 0
#endif
#ifndef HAVE_ASYNC_STORE
#define HAVE_ASYNC_STORE 0
#endif
#ifndef HAVE_WAIT_ASYNC
#define HAVE_WAIT_ASYNC 0
#endif

#define AS1 __attribute__((address_space(1)))
#define AS3 __attribute__((address_space(3)))

typedef int v4i __attribute__((ext_vector_type(4)));

__device__ __forceinline__ void async_load16(const float* g, float* l) {
#if HAVE_ASYNC_LOAD
  __builtin_amdgcn_global_load_async_to_lds_b128(
      (AS1 v4i*)(const_cast<float*>(g)), (AS3 v4i*)l, 0, 0);
#else
  *reinterpret_cast<float4*>(l) = *reinterpret_cast<const float4*>(g);
#endif
}

__device__ __forceinline__ void async_store16(float* g, const float* l) {
#if HAVE_ASYNC_STORE
  __builtin_amdgcn_global_store_async_from_lds_b128(
      (AS1 v4i*)g, (AS3 v4i*)(const_cast<float*>(l)), 0, 0);
#else
  *reinterpret_cast<float4*>(g) = *reinterpret_cast<const float4*>(l);
#endif
}

__device__ __forceinline__ void wait_async() {
#if HAVE_ASYNC_LOAD || HAVE_ASYNC_STORE
#  if HAVE_WAIT_ASYNC
  __builtin_amdgcn_s_wait_asynccnt(0);
#  else
  asm volatile("s_wait_asynccnt 0" ::: "memory");
#  endif
#endif
}

// XOR swizzle on float4 slots: coalesced b128 staging, <=2-way LDS conflicts on
// the strided per-thread reads.
__device__ __forceinline__ int swz(int q)    { return q ^ ((q >> 4) & 15); }
__device__ __forceinline__ int eaddr(int e)  { int q = e >> 2; return (swz(q) << 2) | (e & 3); }

// ---------------- kernel 0: build exact piecewise-linear LUT ----------------
// s(u) = b2 + sum_c w2_c * leaky(w1_c*u + b1_c)
//      = M_j * u + Q_j   on the j-th interval of sorted breakpoints r_c = -b1_c/w1_c
// M_j = 0.6*a1 - 0.4*V + 0.8*P_j ; Q_j = [b2 + 0.6*a0 + 0.4*(W + extra)] - 0.8*W_j
__global__ void k_setup(const float* __restrict__ w1, const float* __restrict__ b1,
                        const float* __restrict__ w2, const float* __restrict__ b2,
                        float* __restrict__ ws) {
  __shared__ float ur[NCH];
  __shared__ float sr[NCH], sv[NCH], svr[NCH];
  __shared__ float red0[NCH], red1[NCH], red2[NCH];
  const int c = threadIdx.x;

  float W1 = w1[c], B1 = b1[c], W2 = w2[c];
  float r, v, extra;
  if (W1 != 0.0f) { r = -B1 / W1; v = W2 * fabsf(W1); extra = 0.0f; }
  else            { r = 3.0e38f;  v = 0.0f;           extra = W2 * fabsf(B1); }

  ur[c] = r;
  red0[c] = W2 * W1;   // -> alpha1
  red1[c] = W2 * B1;   // -> alpha0
  red2[c] = extra;
  __syncthreads();

  // stable rank sort of the 64 breakpoints
  int rank = 0;
  for (int d = 0; d < NCH; ++d) {
    float rd = ur[d];
    rank += (rd < r) || (rd == r && d < c);
  }

  // tree reductions for alpha1 / alpha0 / extra
  for (int s = 32; s > 0; s >>= 1) {
    if (c < s) { red0[c] += red0[c+s]; red1[c] += red1[c+s]; red2[c] += red2[c+s]; }
    __syncthreads();
  }

  sr[rank] = r; sv[rank] = v; svr[rank] = v * r;
  __syncthreads();

  // inclusive prefix sums over sorted v and v*r (Hillis-Steele)
  for (int off = 1; off < NCH; off <<= 1) {
    float a = sv[c], b = svr[c], aa = 0.0f, bb = 0.0f;
    if (c >= off) { aa = sv[c-off]; bb = svr[c-off]; }
    __syncthreads();
    sv[c] = a + aa; svr[c] = b + bb;
    __syncthreads();
  }

  const float alpha1 = red0[0], alpha0 = red1[0], extraS = red2[0];
  const float V = sv[NCH-1], W = svr[NCH-1], B2 = b2[0];
  const float Cst = B2 + 0.6f * alpha0 + 0.4f * (W + extraS);

  ws[WS_R + c] = sr[c];
  for (int j = c; j <= NCH; j += NCH) {
    float P  = (j == 0) ? 0.0f : sv[j-1];
    float Wj = (j == 0) ? 0.0f : svr[j-1];
    ws[WS_MQ + 2*j]     = 0.6f * alpha1 - 0.4f * V + 0.8f * P;
    ws[WS_MQ + 2*j + 1] = Cst - 0.8f * Wj;
  }
}

// per-sample alpha/b evaluation from staged LDS data
__device__ __forceinline__ void eval_ab(const float* sf0, const float* sxx, const float* svm,
                                        const float* lr, const float2* lmq,
                                        float r31, int ea, float& al, float& bv) {
  float f0v = sf0[ea], xv = sxx[ea], mv = svm[ea];
  float u = __builtin_amdgcn_logf(fmaf(f0v, KSCALE, 1.0f));
  int j = (r31 <= u) ? 32 : 0;
  #pragma unroll
  for (int st = 16; st >= 1; st >>= 1)
    if (lr[j + st - 1] <= u) j += st;
  float2 mq = lmq[j];
  float s  = fmaf(mq.x, u, mq.y);
  float e2 = __builtin_amdgcn_exp2f(-LOG2E * s);
  al = 0.98f * mv * __builtin_amdgcn_rcpf(1.0f + e2);
  bv = (1.0f - al) * xv;
}

// ---------------- kernel 1: per-chunk scan aggregates ----------------
__global__ void __launch_bounds__(THREADS)
k_part(const float* __restrict__ x, const float* __restrict__ f0,
       const float* __restrict__ vm, float* __restrict__ ws) {
  extern __shared__ float smem[];
  float* sf0 = smem;
  float* sxx = smem + CHUNK;
  float* svm = smem + 2*CHUNK;
  __shared__ float  lr[NCH];
  __shared__ float2 lmq[NCH+1];
  __shared__ float  wAg[NWAVES], wUg[NWAVES];

  const int tid = threadIdx.x, lane = tid & 31, wave = tid >> 5;
  const int chunk = blockIdx.x;
  const long base = (long)chunk * CHUNK;   // rows are contiguous: row*T + cix*CHUNK

  for (int i = 0; i < F4_PER_THREAD; ++i) {
    int fq = i*THREADS + tid, dq = swz(fq);
    async_load16(f0 + base + 4l*fq, sf0 + 4*dq);
    async_load16(x  + base + 4l*fq, sxx + 4*dq);
    async_load16(vm + base + 4l*fq, svm + 4*dq);
  }
  if (tid < NCH)   lr[tid]  = ws[WS_R + tid];
  if (tid < NCH+1) lmq[tid] = reinterpret_cast<const float2*>(ws + WS_MQ)[tid];
  wait_async();
  __syncthreads();

  const float r31 = lr[31];
  float A = 1.0f, U = 0.0f;
  #pragma unroll
  for (int k = 0; k < SPT; ++k) {
    int ea = eaddr(tid*SPT + k);
    float al, bv;
    eval_ab(sf0, sxx, svm, lr, lmq, r31, ea, al, bv);
    U = fmaf(al, U, bv);
    A *= al;
  }

  // wave32 inclusive scan (combine(prev,cur) = (aP*aC, aC*uP + uC))
  for (int off = 1; off < 32; off <<= 1) {
    float pA = __shfl_up(A, off, 32), pU = __shfl_up(U, off, 32);
    if (lane >= off) { U = fmaf(A, pU, U); A *= pA; }
  }
  if (lane == 31) { wAg[wave] = A; wUg[wave] = U; }
  __syncthreads();
  if (tid == 0) {
    float pa = 1.0f, pu = 0.0f;
    for (int w = 0; w < NWAVES; ++w) {
      float ca = wAg[w], cu = wUg[w];
      pu = fmaf(ca, pu, cu); pa *= ca;
    }
    ws[WS_AGG + 2*chunk]     = pa;
    ws[WS_AGG + 2*chunk + 1] = pu;
  }
}

// ---------------- kernel 2: spine scan over chunk aggregates ----------------
__global__ void k_spine(const float* __restrict__ init, float* __restrict__ ws) {
  int row = threadIdx.x;
  if (row >= BATCH) return;
  float y = init[row];
  for (int j = 0; j < CPR; ++j) {
    int cid = row*CPR + j;
    ws[WS_CARRY + cid] = y;                              // exclusive carry-in
    y = fmaf(ws[WS_AGG + 2*cid], y, ws[WS_AGG + 2*cid + 1]);
  }
}

// ---------------- kernel 3: apply carries, emit y ----------------
__global__ void __launch_bounds__(THREADS)
k_final(const float* __restrict__ x, const float* __restrict__ f0,
        const float* __restrict__ vm, const float* __restrict__ ws,
        float* __restrict__ out) {
  extern __shared__ float smem[];
  float* sf0 = smem;            // f0 -> alpha (in place)
  float* sxx = smem + CHUNK;    // x  -> b     (in place)
  float* svm = smem + 2*CHUNK;  // mask -> y   (in place)
  __shared__ float  lr[NCH];
  __shared__ float2 lmq[NCH+1];
  __shared__ float  wAg[NWAVES], wUg[NWAVES];

  const int tid = threadIdx.x, lane = tid & 31, wave = tid >> 5;
  const int chunk = blockIdx.x;
  const long base = (long)chunk * CHUNK;

  for (int i = 0; i < F4_PER_THREAD; ++i) {
    int fq = i*THREADS + tid, dq = swz(fq);
    async_load16(f0 + base + 4l*fq, sf0 + 4*dq);
    async_load16(x  + base + 4l*fq, sxx + 4*dq);
    async_load16(vm + base + 4l*fq, svm + 4*dq);
  }
  if (tid < NCH)   lr[tid]  = ws[WS_R + tid];
  if (tid < NCH+1) lmq[tid] = reinterpret_cast<const float2*>(ws + WS_MQ)[tid];
  wait_async();
  __syncthreads();

  const float r31 = lr[31];
  float A = 1.0f, U = 0.0f;
  #pragma unroll
  for (int k = 0; k < SPT; ++k) {
    int ea = eaddr(tid*SPT + k);
    float al, bv;
    eval_ab(sf0, sxx, svm, lr, lmq, r31, ea, al, bv);
    U = fmaf(al, U, bv);
    A *= al;
    sf0[ea] = al;   // stash per-element alpha / b in place (thread-owned slots)
    sxx[ea] = bv;
  }

  // block-exclusive scan of thread aggregates
  for (int off = 1; off < 32; off <<= 1) {
    float pA = __shfl_up(A, off, 32), pU = __shfl_up(U, off, 32);
    if (lane >= off) { U = fmaf(A, pU, U); A *= pA; }
  }
  if (lane == 31) { wAg[wave] = A; wUg[wave] = U; }
  __syncthreads();
  if (tid == 0) {
    float pa = 1.0f, pu = 0.0f;
    for (int w = 0; w < NWAVES; ++w) {
      float ca = wAg[w], cu = wUg[w];
      wAg[w] = pa; wUg[w] = pu;            // exclusive wave prefixes
      pu = fmaf(ca, pu, cu); pa *= ca;
    }
  }
  __syncthreads();

  float la = __shfl_up(A, 1, 32), lu = __shfl_up(U, 1, 32);
  if (lane == 0) { la = 1.0f; lu = 0.0f; }
  const float Ea = wAg[wave], Eu = wUg[wave];
  float y = fmaf(Ea * la, ws[WS_CARRY + chunk], fmaf(la, Eu, lu));

  #pragma unroll
  for (int k = 0; k < SPT; ++k) {
    int ea = eaddr(tid*SPT + k);
    y = fmaf(sf0[ea], y, sxx[ea]);
    svm[ea] = y;
  }
  __syncthreads();

  for (int i = 0; i < F4_PER_THREAD; ++i) {
    int fq = i*THREADS + tid, dq = swz(fq);
    async_store16(out + base + 4l*fq, svm + 4*dq);
  }
  // S_ENDPGM performs an implicit wait-idle, draining outstanding async stores.
}

// ---------------- host entry ----------------
extern "C" void kernel_launch(void* const* d_in, const int* in_sizes, int n_in,
                              void* d_out, int out_size, void* d_ws, size_t ws_size,
                              hipStream_t stream) {
  const float* x    = (const float*)d_in[0];
  const float* f0   = (const float*)d_in[1];
  const float* vm   = (const float*)d_in[2];
  const float* init = (const float*)d_in[3];
  const float* w1   = (const float*)d_in[4];
  const float* b1   = (const float*)d_in[5];
  const float* w2   = (const float*)d_in[6];
  const float* b2   = (const float*)d_in[7];
  float* out = (float*)d_out;
  float* ws  = (float*)d_ws;   // uses < 32 KB

  const size_t dyn = 3 * CHUNK * sizeof(float);  // 48 KB staged LDS per block
  k_setup<<<dim3(1),      dim3(NCH),     0,   stream>>>(w1, b1, w2, b2, ws);
  k_part <<<dim3(NCHUNK), dim3(THREADS), dyn, stream>>>(x, f0, vm, ws);
  k_spine<<<dim3(1),      dim3(32),      0,   stream>>>(init, ws);
  k_final<<<dim3(NCHUNK), dim3(THREADS), dyn, stream>>>(x, f0, vm, ws, out);
}